// NeuralBondOrder_64407329571244
// MI455X (gfx1250) — compile-verified
//
#include <hip/hip_runtime.h>

typedef float v2f __attribute__((ext_vector_type(2)));
typedef float v8f __attribute__((ext_vector_type(8)));

#define HID 64

// ---------------------------------------------------------------- zero output
__global__ void zero_kernel(float* out, int n) {
  int i = blockIdx.x * blockDim.x + threadIdx.x;
  if (i < n) out[i] = 0.0f;
}

// ------------------------------------------------- phase 1: node projections
// Per 16-node tile: accG = x_tile @ [Wsg | Wdg | 0...]   (cols 0,1)
//                   accP = x0_tile @ [Wsp | Wdp | 0...]  (cols 0..7)
// Outputs: gsA[node] = x.Wsg + bsg + beg   (beg folded here)
//          gdA[node] = x.Wdg + bdg
//          peA[node*8 + c] = exp(x0.Wsp + bsp)[c] (c<4), exp(x0.Wdp)[c-4] (c>=4)
// B matrices are staged in LDS as [n][k] so the K-loop reads them with plain
// ds_load_b64 (no divergent global loads -> no saveexec branching).
__global__ __launch_bounds__(256) void node_kernel(
    const float* __restrict__ x,   const float* __restrict__ x0,
    const float* __restrict__ Wsg, const float* __restrict__ bsg,
    const float* __restrict__ Wdg, const float* __restrict__ bdg,
    const float* __restrict__ beg,
    const float* __restrict__ Wsp, const float* __restrict__ bsp,
    const float* __restrict__ Wdp,
    float* __restrict__ gsA, float* __restrict__ gdA, float* __restrict__ peA,
    int N) {
  __shared__ float bgL[16 * HID];   // [n][k]: col0=Wsg, col1=Wdg, rest 0
  __shared__ float bpL[16 * HID];   // [n][k]: cols0-3=Wsp, cols4-7=Wdp, rest 0
  const int tid = threadIdx.x;

  // Branch-free staging: all global loads unconditional (k<64 always valid),
  // lane selection lowers to v_cndmask.
  for (int idx = tid; idx < 16 * HID; idx += 256) {
    const int n = idx >> 6, k = idx & 63;
    const float vs  = Wsg[k];
    const float vd  = Wdg[k];
    const float vsp = Wsp[k * 4 + (n & 3)];
    const float vdp = Wdp[k * 4 + (n & 3)];
    bgL[idx] = (n == 0) ? vs : (n == 1) ? vd : 0.0f;
    bpL[idx] = (n < 4) ? vsp : (n < 8) ? vdp : 0.0f;
  }
  __syncthreads();

  const int lane = tid & 31;
  const int wv   = tid >> 5;
  const int tile = (blockIdx.x * 8 + wv) * 16;
  if (tile < N) {                        // wave-uniform guard (no early return)
    const int n  = lane & 15;            // N-column of this lane (C/D layout)
    const int hi = lane >> 4;            // K half-select for A/B striping
    int nodeA = tile + n; if (nodeA >= N) nodeA = N - 1;   // clamped A row

    v8f accG = {0,0,0,0,0,0,0,0};
    v8f accP = {0,0,0,0,0,0,0,0};
#pragma unroll
    for (int k0 = 0; k0 < HID; k0 += 4) {
      const int ka = k0 + hi * 2;        // VGPR0 holds K=k0 / k0+2, VGPR1 +1
      v2f ag = *(const v2f*)(x  + (size_t)nodeA * HID + ka);
      v2f ap = *(const v2f*)(x0 + (size_t)nodeA * HID + ka);
      v2f bg = *(const v2f*)(&bgL[n * HID + ka]);
      v2f bp = *(const v2f*)(&bpL[n * HID + ka]);
      accG = __builtin_amdgcn_wmma_f32_16x16x4_f32(false, ag, false, bg, (short)0, accG, false, false);
      accP = __builtin_amdgcn_wmma_f32_16x16x4_f32(false, ap, false, bp, (short)0, accP, false, false);
    }

    const float cg = (n == 0) ? (bsg[0] + beg[0]) : (n == 1) ? bdg[0] : 0.0f;
    const float cp = (n < 4) ? bsp[n] : 0.0f;
#pragma unroll
    for (int r = 0; r < 8; ++r) {
      const int node = tile + r + hi * 8;  // D: VGPR r = M=r (lo) / M=r+8 (hi)
      if (node < N) {
        if (n == 0) gsA[node] = accG[r] + cg;
        if (n == 1) gdA[node] = accG[r] + cg;
        if (n < 8)  peA[(size_t)node * 8 + n] = __expf(accP[r] + cp);
      }
    }
  }
}

// --------------------------------------- phase 2: edge pass (the 410MB loop)
// Each wave: 32 edges = two 16-edge WMMA chains computing y_row . Weg,
// Weg replicated across all 16 B columns so every lane's acc holds valid dots.
// y is streamed once -> non-temporal loads keep L2 free for the gather tables.
__global__ __launch_bounds__(256) void edge_kernel(
    const float* __restrict__ y,   const float* __restrict__ bond,
    const int*   __restrict__ src, const int*   __restrict__ dst,
    const float* __restrict__ Weg,
    const float* __restrict__ gsA, const float* __restrict__ gdA,
    const float* __restrict__ peA,
    float* __restrict__ atom, int E) {
  __shared__ float dots[8][32];
  const int lane = threadIdx.x & 31;
  const int wv   = threadIdx.x >> 5;
  const long e0  = ((long)blockIdx.x * 8 + wv) * 32;
  const int m    = lane & 15;
  const int hi   = lane >> 4;

  long eR0 = e0 + m;       if (eR0 >= E) eR0 = E - 1;
  long eR1 = e0 + 16 + m;  if (eR1 >= E) eR1 = E - 1;
  const float* row0 = y + eR0 * HID;
  const float* row1 = y + eR1 * HID;
  __builtin_prefetch(row0, 0, 3); __builtin_prefetch(row0 + 32, 0, 3);
  __builtin_prefetch(row1, 0, 3); __builtin_prefetch(row1 + 32, 0, 3);

  // Hoist the 16 Weg chunks (K striped exactly like A: ka = 4*c + hi*2).
  v2f wB[16];
#pragma unroll
  for (int c = 0; c < 16; ++c) wB[c] = *(const v2f*)(Weg + c * 4 + hi * 2);

  v8f acc0 = {0,0,0,0,0,0,0,0};
  v8f acc1 = {0,0,0,0,0,0,0,0};
#pragma unroll
  for (int c = 0; c < 16; ++c) {
    const int ka = c * 4 + hi * 2;
    v2f a0 = __builtin_nontemporal_load((const v2f*)(row0 + ka));
    v2f a1 = __builtin_nontemporal_load((const v2f*)(row1 + ka));
    acc0 = __builtin_amdgcn_wmma_f32_16x16x4_f32(false, a0, false, wB[c], (short)0, acc0, false, false);
    acc1 = __builtin_amdgcn_wmma_f32_16x16x4_f32(false, a1, false, wB[c], (short)0, acc1, false, false);
  }

  // lane 0 holds dots for edges M=0..7 in acc[0..7]; lane 16 for M=8..15
  if (lane == 0) {
#pragma unroll
    for (int r = 0; r < 8; ++r) { dots[wv][r]     = acc0[r]; dots[wv][16 + r] = acc1[r]; }
  } else if (lane == 16) {
#pragma unroll
    for (int r = 0; r < 8; ++r) { dots[wv][8 + r] = acc0[r]; dots[wv][24 + r] = acc1[r]; }
  }
  __syncthreads();

  const long e = e0 + lane;
  if (e < E) {
    const float ydot = dots[wv][lane];
    const int se = src[e], de = dst[e];
    const float mm = gsA[se] + gdA[de] + ydot;             // biases folded
    const float bo = 1.0f / (1.0f + __expf(-mm));          // sigmoid
    const float r  = bond[e];
    const float4 ps = *(const float4*)(peA + (size_t)se * 8);      // exp(p_src)
    const float4 pd = *(const float4*)(peA + (size_t)de * 8 + 4);  // exp(p_dst)
    const float P0 = ps.x * pd.x, P1 = ps.y * pd.y;
    const float P2 = ps.z * pd.z, P3 = ps.w * pd.w;
    float c;
    if      (r > 4.0f) c = 0.0f;
    else if (r < 3.8f) c = 1.0f;
    else c = 0.5f - 0.5f * __sinf(3.14159265358979f * (r - 3.9f) * 5.0f);
    if (c != 0.0f) {
      const float frep = P0 * __expf(-P1 * r);
      const float fatt = P2 * __expf(-P3 * r);
      atomicAdd(&atom[de], c * (frep - bo * fatt));
    }
  }
}

// ---------------------------------------------------- phase 3: mean readout
__global__ void reduce_kernel(float* out, int N) {
  __shared__ float s[256];
  float sum = 0.0f;
  for (int i = threadIdx.x; i < N; i += 256) sum += out[1 + i];
  s[threadIdx.x] = sum;
  __syncthreads();
  for (int st = 128; st > 0; st >>= 1) {
    if ((int)threadIdx.x < st) s[threadIdx.x] += s[threadIdx.x + st];
    __syncthreads();
  }
  if (threadIdx.x == 0) out[0] = s[0] / (float)N;
}

// ---------------------------------------------------------------------------
extern "C" void kernel_launch(void* const* d_in, const int* in_sizes, int n_in,
                              void* d_out, int out_size, void* d_ws, size_t ws_size,
                              hipStream_t stream) {
  const float* x   = (const float*)d_in[0];
  const float* x0  = (const float*)d_in[1];
  const float* y   = (const float*)d_in[2];
  const float* bl  = (const float*)d_in[3];
  const int*   src = (const int*)  d_in[4];
  const int*   dst = (const int*)  d_in[5];
  const float* Wsg = (const float*)d_in[6];
  const float* bsg = (const float*)d_in[7];
  const float* Wdg = (const float*)d_in[8];
  const float* bdg = (const float*)d_in[9];
  const float* Weg = (const float*)d_in[10];
  const float* beg = (const float*)d_in[11];
  const float* Wsp = (const float*)d_in[12];
  const float* bsp = (const float*)d_in[13];
  const float* Wdp = (const float*)d_in[14];

  const int N = in_sizes[0] / HID;      // 50000
  const int E = in_sizes[3];            // 1600000

  float* out = (float*)d_out;           // [0]=energy, [1..N]=atomwise
  float* gsA = (float*)d_ws;            // N floats
  float* gdA = gsA + N;                 // N floats
  float* peA = gdA + N;                 // 8*N floats (exp'd pair params)

  zero_kernel<<<(out_size + 255) / 256, 256, 0, stream>>>(out, out_size);

  const int ntiles = (N + 15) / 16;
  node_kernel<<<(ntiles + 7) / 8, 256, 0, stream>>>(
      x, x0, Wsg, bsg, Wdg, bdg, beg, Wsp, bsp, Wdp, gsA, gdA, peA, N);

  const int ewaves = (E + 31) / 32;
  edge_kernel<<<(ewaves + 7) / 8, 256, 0, stream>>>(
      y, bl, src, dst, Weg, gsA, gdA, peA, out + 1, E);

  reduce_kernel<<<1, 256, 0, stream>>>(out, N);
}